// LSTM_62414464745738
// MI455X (gfx1250) — compile-verified
//
#include <hip/hip_runtime.h>
#include <math.h>

#define LSTM_SEQ    512
#define LSTM_BATCH  64
#define LSTM_INPUT  1024
#define LSTM_HIDDEN 1024

typedef __attribute__((ext_vector_type(2))) float v2f;
typedef __attribute__((ext_vector_type(8))) float v8f;

__device__ __forceinline__ float sigmoidf_(float x) {
    return 1.0f / (1.0f + __expf(-x));
}

// One fused LSTM timestep for one layer:
//   gates = x_t @ W_ih^T + h_{t-1} @ W_hh^T + (b_ih + b_hh)
//   c_t   = sig(f)*c_{t-1} + sig(i)*tanh(g) ;  h_t = sig(o)*tanh(c_t)
// grid = (HIDDEN/32, BATCH/16), block = 128 = 4 wave32s.
// Wave g computes gate g's 16x32 tile as two 16x16 WMMA accumulators that
// share one A fragment per K-quad (3 loads : 2 v_wmma_f32_16x16x4_f32).
__global__ void lstm_step_wmma_f32(
    const float* __restrict__ xin,    // [BATCH, INPUT] layer input at step t
    const float* __restrict__ hprev,  // [BATCH, HIDDEN] (unused when first_step)
    const float* __restrict__ w_ih,   // [4H, INPUT]
    const float* __restrict__ w_hh,   // [4H, HIDDEN]
    const float* __restrict__ b_ih,   // [4H]
    const float* __restrict__ b_hh,   // [4H]
    float* __restrict__ cstate,       // [BATCH, HIDDEN] updated in place
    float* __restrict__ hout,         // [BATCH, HIDDEN]
    int first_step)
{
    __shared__ float gtile[4][16][32];   // [gate][m][n]  (8 KB)

    const int tid     = threadIdx.x;
    const int wave    = tid >> 5;        // 0..3 == gate index (f,i,g,o)
    const int lane    = tid & 31;
    const int r       = lane & 15;       // row-within-halfwave selector
    const int halfsel = lane >> 4;       // 0: K pair {k,k+1}; 1: {k+2,k+3}
    const int koff    = 2 * halfsel;

    const int n0 = blockIdx.x * 32;      // hidden-column tile (32 wide)
    const int m0 = blockIdx.y * 16;      // batch-row tile

    // Per-lane fragment pointers (ISA 16x4 A / 4x16 B layouts: each lane holds
    // one contiguous float2 of its row at K offset 2*halfsel).
    const float* arow_x  = xin  + (size_t)(m0 + r) * LSTM_INPUT;
    const float* brow_i0 = w_ih + (size_t)(wave * LSTM_HIDDEN + n0 + r)      * LSTM_INPUT;
    const float* brow_i1 = w_ih + (size_t)(wave * LSTM_HIDDEN + n0 + 16 + r) * LSTM_INPUT;

    v8f acc0 = {0.f, 0.f, 0.f, 0.f, 0.f, 0.f, 0.f, 0.f};
    v8f acc1 = {0.f, 0.f, 0.f, 0.f, 0.f, 0.f, 0.f, 0.f};

    // ---- K loop over input contribution: x_t @ W_ih^T ----
#pragma unroll 8
    for (int k = 0; k < LSTM_INPUT; k += 4) {
        v2f a  = *(const v2f*)(arow_x  + k + koff);
        v2f b0 = *(const v2f*)(brow_i0 + k + koff);
        v2f b1 = *(const v2f*)(brow_i1 + k + koff);
        acc0 = __builtin_amdgcn_wmma_f32_16x16x4_f32(
            false, a, false, b0, (short)0, acc0, false, false);
        acc1 = __builtin_amdgcn_wmma_f32_16x16x4_f32(
            false, a, false, b1, (short)0, acc1, false, false);
    }

    // ---- K loop over recurrent contribution: h_{t-1} @ W_hh^T ----
    if (!first_step) {   // uniform branch: EXEC stays all-ones for WMMA
        const float* arow_h  = hprev + (size_t)(m0 + r) * LSTM_HIDDEN;
        const float* brow_h0 = w_hh + (size_t)(wave * LSTM_HIDDEN + n0 + r)      * LSTM_HIDDEN;
        const float* brow_h1 = w_hh + (size_t)(wave * LSTM_HIDDEN + n0 + 16 + r) * LSTM_HIDDEN;
#pragma unroll 8
        for (int k = 0; k < LSTM_HIDDEN; k += 4) {
            v2f a  = *(const v2f*)(arow_h  + k + koff);
            v2f b0 = *(const v2f*)(brow_h0 + k + koff);
            v2f b1 = *(const v2f*)(brow_h1 + k + koff);
            acc0 = __builtin_amdgcn_wmma_f32_16x16x4_f32(
                false, a, false, b0, (short)0, acc0, false, false);
            acc1 = __builtin_amdgcn_wmma_f32_16x16x4_f32(
                false, a, false, b1, (short)0, acc1, false, false);
        }
    }

    // ---- Spill gate tiles to LDS ----
    // C layout: lane (r, halfsel), VGPR v -> element (M = v + 8*halfsel, N = r)
#pragma unroll
    for (int v = 0; v < 8; ++v) {
        gtile[wave][v + 8 * halfsel][r]      = acc0[v];
        gtile[wave][v + 8 * halfsel][16 + r] = acc1[v];
    }
    __syncthreads();

    // ---- Fused elementwise epilogue ----
    // 512 elements over 128 threads: each thread owns one hidden column and 4
    // batch rows, so the per-gate bias sum is loaded once per thread.
    {
        const int ni    = tid & 31;           // column within tile
        const int mbase = (tid >> 5) * 4;     // 4 rows per thread
        const int col   = n0 + ni;            // hidden unit index

        const float fb = b_ih[0 * LSTM_HIDDEN + col] + b_hh[0 * LSTM_HIDDEN + col];
        const float ib = b_ih[1 * LSTM_HIDDEN + col] + b_hh[1 * LSTM_HIDDEN + col];
        const float gb = b_ih[2 * LSTM_HIDDEN + col] + b_hh[2 * LSTM_HIDDEN + col];
        const float ob = b_ih[3 * LSTM_HIDDEN + col] + b_hh[3 * LSTM_HIDDEN + col];

#pragma unroll
        for (int dm = 0; dm < 4; ++dm) {
            const int mi = mbase + dm;
            float fg = sigmoidf_(gtile[0][mi][ni] + fb);
            float ig = sigmoidf_(gtile[1][mi][ni] + ib);
            float gg = tanhf    (gtile[2][mi][ni] + gb);
            float og = sigmoidf_(gtile[3][mi][ni] + ob);

            const size_t off = (size_t)(m0 + mi) * LSTM_HIDDEN + col;
            const float cprev = first_step ? 0.0f : cstate[off];
            const float cnew  = fg * cprev + ig * gg;
            cstate[off] = cnew;
            hout[off]   = og * tanhf(cnew);
        }
    }
}

extern "C" void kernel_launch(void* const* d_in, const int* in_sizes, int n_in,
                              void* d_out, int out_size, void* d_ws, size_t ws_size,
                              hipStream_t stream) {
    (void)in_sizes; (void)n_in; (void)out_size; (void)ws_size;

    const float* x    = (const float*)d_in[0];   // [SEQ, BATCH, INPUT]
    const float* w_ih = (const float*)d_in[1];   // [4H, INPUT]
    const float* w_hh = (const float*)d_in[2];   // [4H, HIDDEN]
    const float* b_ih = (const float*)d_in[3];   // [4H]
    const float* b_hh = (const float*)d_in[4];   // [4H]
    float* out = (float*)d_out;                  // [h1,h2,c1,c2] each [64,1024]

    const size_t BH = (size_t)LSTM_BATCH * LSTM_HIDDEN;   // 65536 elements
    float* ws = (float*)d_ws;
    float* c1    = ws + 0 * BH;
    float* c2    = ws + 1 * BH;
    float* h1[2] = { ws + 2 * BH, ws + 3 * BH };
    float* h2[2] = { ws + 4 * BH, ws + 5 * BH };

    dim3 grid(LSTM_HIDDEN / 32, LSTM_BATCH / 16);   // 32 x 4 = 128 blocks
    dim3 block(128);                                 // 4 wave32s (one per gate)

    for (int t = 0; t < LSTM_SEQ; ++t) {
        const int wr = t & 1;
        const int rd = wr ^ 1;
        // Layer 1 (reads x_t, h1[rd]; writes c1, h1[wr])
        lstm_step_wmma_f32<<<grid, block, 0, stream>>>(
            x + (size_t)t * LSTM_BATCH * LSTM_INPUT, h1[rd],
            w_ih, w_hh, b_ih, b_hh, c1, h1[wr], t == 0 ? 1 : 0);
        // Layer 2 (reads layer-1 output h1[wr], h2[rd]; writes c2, h2[wr])
        lstm_step_wmma_f32<<<grid, block, 0, stream>>>(
            h1[wr], h2[rd],
            w_ih, w_hh, b_ih, b_hh, c2, h2[wr], t == 0 ? 1 : 0);
    }

    const int last = (LSTM_SEQ - 1) & 1;   // == 1
    // Output layout: h_n[0], h_n[1], c_n[0], c_n[1]
    hipMemcpyAsync(out + 0 * BH, h1[last], BH * sizeof(float),
                   hipMemcpyDeviceToDevice, stream);
    hipMemcpyAsync(out + 1 * BH, h2[last], BH * sizeof(float),
                   hipMemcpyDeviceToDevice, stream);
    hipMemcpyAsync(out + 2 * BH, c1, BH * sizeof(float),
                   hipMemcpyDeviceToDevice, stream);
    hipMemcpyAsync(out + 3 * BH, c2, BH * sizeof(float),
                   hipMemcpyDeviceToDevice, stream);
}